// IDWTLayer_24532853195399
// MI455X (gfx1250) — compile-verified
//
#include <hip/hip_runtime.h>
#include <cstdint>

// ---------------------------------------------------------------------------
// Inverse Haar DWT (conv2d_transpose k=2 s=2), NHWC.
//   in : 4 x [8,256,256,32] f32   out: [8,512,512,32] f32
// Memory-bound: 512 MiB traffic @ 23.3 TB/s ~ 23 us. Strategy:
//   * b128 (float4) loads staged via CDNA5 GLOBAL_LOAD_ASYNC_TO_LDS_B128
//     (ASYNCcnt-tracked), 2-deep ping-pong software pipeline per wave.
//   * butterfly in VALU (trivial vs BW), non-temporal b128 stores.
// ---------------------------------------------------------------------------

typedef float v4f __attribute__((ext_vector_type(4)));

static constexpr unsigned BLOCK = 256;   // 8 waves (wave32)
static constexpr unsigned NITER = 4;     // pipelined tiles per block
static constexpr unsigned WP    = 256;   // coefficient width
static constexpr unsigned CCH   = 32;    // channels
static constexpr unsigned OROW  = 2u * WP * CCH;   // 16384 floats per output row

// One 16-byte per-lane async copy: copies 16 bytes from global address g to the LDS address in lds_addr
__device__ __forceinline__ void async_ld16(uint32_t lds_addr, const float* g) {
    asm volatile("global_load_async_to_lds_b128 %0, %1, off"
                 :: "v"(lds_addr), "v"(g)
                 : "memory");
}

// Butterfly + interleaved 2x2 store for one (pixel, channel-quad).
//   gq = global quad index; p = gq>>3 pixel, q = gq&7 channel quad.
__device__ __forceinline__ void haar_store(float* __restrict__ out, unsigned gq,
                                           v4f A, v4f Hc, v4f Vc, v4f Dc) {
    const unsigned p = gq >> 3;
    const unsigned q = gq & 7u;
    const unsigned j = p & (WP - 1u);     // column
    const unsigned r = p >> 8;            // b*256 + i  (row pair index)

    // output row 2i has [o00 ch0..31 | o01 ch0..31] contiguous; row 2i+1 same for o10/o11
    const size_t row0 = (size_t)(2u * r) * OROW + (size_t)(2u * j) * CCH + (size_t)q * 4u;
    const size_t row1 = row0 + OROW;

    const v4f P = 0.5f * (A + Vc);
    const v4f M = 0.5f * (A - Vc);
    const v4f Q = 0.5f * (Hc - Dc);
    const v4f o00 = P + Q;   // 0.5*(A+H+V-D)
    const v4f o01 = P - Q;   // 0.5*(A-H+V+D)
    const v4f o10 = M + Q;   // 0.5*(A+H-V-D)
    const v4f o11 = M - Q;   // 0.5*(A-H-V+D)

    __builtin_nontemporal_store(o00, (v4f*)(out + row0));
    __builtin_nontemporal_store(o01, (v4f*)(out + row0 + CCH));
    __builtin_nontemporal_store(o10, (v4f*)(out + row1));
    __builtin_nontemporal_store(o11, (v4f*)(out + row1 + CCH));
}

__global__ __launch_bounds__(BLOCK) void idwt_haar_async(
    const float* __restrict__ cA, const float* __restrict__ cH,
    const float* __restrict__ cV, const float* __restrict__ cD,
    float* __restrict__ out)
{
    // [buffer][input][lane-slot]: 2*4*256*16B = 32 KB. Each wave only touches
    // its own 32 slots -> no inter-wave sharing, no barriers needed.
    __shared__ v4f sbuf[2][4][BLOCK];

    const unsigned tid  = threadIdx.x;
    const unsigned base = blockIdx.x * (BLOCK * NITER) + tid;   // quad index, iter 0

    auto issue = [&](unsigned it) {
        const unsigned gq  = base + it * BLOCK;
        const unsigned buf = it & 1u;
        const size_t   off = (size_t)gq * 4u;   // input is a flat float4 stream
        async_ld16((uint32_t)(uintptr_t)&sbuf[buf][0][tid], cA + off);
        async_ld16((uint32_t)(uintptr_t)&sbuf[buf][1][tid], cH + off);
        async_ld16((uint32_t)(uintptr_t)&sbuf[buf][2][tid], cV + off);
        async_ld16((uint32_t)(uintptr_t)&sbuf[buf][3][tid], cD + off);
    };

    issue(0);
#pragma unroll
    for (unsigned it = 0; it < NITER; ++it) {
        const unsigned buf = it & 1u;
        if (it + 1u < NITER) {
            issue(it + 1u);                                     // 4 newer in flight
            asm volatile("s_wait_asynccnt 4" ::: "memory");     // retire current tile
        } else {
            asm volatile("s_wait_asynccnt 0" ::: "memory");
        }
        const v4f A  = sbuf[buf][0][tid];
        const v4f Hc = sbuf[buf][1][tid];
        const v4f Vc = sbuf[buf][2][tid];
        const v4f Dc = sbuf[buf][3][tid];
        haar_store(out, base + it * BLOCK, A, Hc, Vc, Dc);
    }
}

// Direct-load tail (only used if quad count isn't a multiple of BLOCK*NITER;
// for the reference shape it never launches, kept for robustness).
__global__ __launch_bounds__(BLOCK) void idwt_haar_tail(
    const float* __restrict__ cA, const float* __restrict__ cH,
    const float* __restrict__ cV, const float* __restrict__ cD,
    float* __restrict__ out, unsigned startQuad, unsigned nQuads)
{
    const unsigned gq = startQuad + blockIdx.x * BLOCK + threadIdx.x;
    if (gq >= nQuads) return;
    const size_t off = (size_t)gq * 4u;
    const v4f A  = __builtin_nontemporal_load((const v4f*)(cA + off));
    const v4f Hc = __builtin_nontemporal_load((const v4f*)(cH + off));
    const v4f Vc = __builtin_nontemporal_load((const v4f*)(cV + off));
    const v4f Dc = __builtin_nontemporal_load((const v4f*)(cD + off));
    haar_store(out, gq, A, Hc, Vc, Dc);
}

extern "C" void kernel_launch(void* const* d_in, const int* in_sizes, int n_in,
                              void* d_out, int out_size, void* d_ws, size_t ws_size,
                              hipStream_t stream) {
    (void)n_in; (void)out_size; (void)d_ws; (void)ws_size;
    const float* cA = (const float*)d_in[0];
    const float* cH = (const float*)d_in[1];
    const float* cV = (const float*)d_in[2];
    const float* cD = (const float*)d_in[3];
    float* out = (float*)d_out;

    const unsigned n      = (unsigned)in_sizes[0];   // 16,777,216 floats per input
    const unsigned nQuads = n / 4u;                  // 4,194,304 float4 quads
    const unsigned perBlk = BLOCK * NITER;           // 1024 quads per block
    const unsigned grid   = nQuads / perBlk;         // 4096 blocks (exact)

    if (grid)
        idwt_haar_async<<<grid, BLOCK, 0, stream>>>(cA, cH, cV, cD, out);

    const unsigned done = grid * perBlk;
    if (done < nQuads) {
        const unsigned rem = nQuads - done;
        idwt_haar_tail<<<(rem + BLOCK - 1) / BLOCK, BLOCK, 0, stream>>>(
            cA, cH, cV, cD, out, done, nQuads);
    }
}